// MessagePassingNet_58308476010700
// MI455X (gfx1250) — compile-verified
//
#include <hip/hip_runtime.h>
#include <math.h>

#define N_NODES 25000
#define N_EDGES 400000
#define FN 32
#define FE 16
#define HDIM 16
#define MDIM 16
#define N_ITERS 3

typedef float v2f __attribute__((ext_vector_type(2)));
typedef float v8f __attribute__((ext_vector_type(8)));

static __device__ __forceinline__ v8f wmma_f32_4(v2f a, v2f b, v8f c) {
  // D = A(16x4 f32) * B(4x16 f32) + C(16x16 f32)
  return __builtin_amdgcn_wmma_f32_16x16x4_f32(
      /*neg_a=*/false, a, /*neg_b=*/false, b,
      /*c_mod=*/(short)0, c, /*reuse_a=*/false, /*reuse_b=*/false);
}

// ---------------------------------------------------------------------------
// Kernel 0: generic zero fill
// ---------------------------------------------------------------------------
__global__ void zero_kernel(float* __restrict__ p, int n) {
  int i = blockIdx.x * blockDim.x + threadIdx.x;
  if (i < n) p[i] = 0.0f;
}

// ---------------------------------------------------------------------------
// Kernel 1: hidden0 = node_features @ W_init + b_init   (WMMA f32 16x16x4)
// One wave per 16-node tile. K = FN = 32 -> 8 k-slices.
// A-operand K sub-offset (2*half) is baked into the LOAD ADDRESS so every
// register index is compile-time (no cndmask select chains).
// ---------------------------------------------------------------------------
__global__ __launch_bounds__(32) void init_hidden_kernel(
    const float* __restrict__ nf, const float* __restrict__ Wini,
    const float* __restrict__ bini, float* __restrict__ hidden,
    float* __restrict__ hidden0) {
  const unsigned lane = threadIdx.x & 31u;
  const unsigned col  = lane & 15u;   // N-column (and A-row M)
  const unsigned half = lane >> 4;    // K sub-offset selector
  const int row0 = blockIdx.x * 16;

  int m = row0 + (int)col;
  if (m >= N_NODES) m = N_NODES - 1;  // clamp (dup loads are harmless)

  // aq[ks] = nf[m, 4*ks + 2*half + {0,1}]
  const float* nb = nf + (size_t)m * FN + 2 * half;
  float2 aq[FN / 4];
#pragma unroll
  for (int q = 0; q < FN / 4; ++q) aq[q] = *(const float2*)(nb + 4 * q);

  // B columns for this lane: Wini[k, col] with k = 4*ks + 2*half + {0,1}
  const float* wb = Wini + (size_t)(2 * half) * HDIM + col;

  const float bcol = bini[col];
  v8f c;
#pragma unroll
  for (int r = 0; r < 8; ++r) c[r] = bcol;

#pragma unroll
  for (int ks = 0; ks < FN / 4; ++ks) {
    v2f a, b;
    a.x = aq[ks].x;
    a.y = aq[ks].y;
    b.x = wb[(4 * ks + 0) * HDIM];
    b.y = wb[(4 * ks + 1) * HDIM];
    c = wmma_f32_4(a, b, c);
  }

#pragma unroll
  for (int r = 0; r < 8; ++r) {
    const int row = row0 + r + 8 * (int)half;
    if (row < N_NODES) {
      hidden [(size_t)row * HDIM + col] = c[r];
      hidden0[(size_t)row * HDIM + col] = c[r];
    }
  }
}

// ---------------------------------------------------------------------------
// Kernel 2: per-edge messages + scatter-add (recompute edge transform)
//   msg[e,m] = sum_{f,h} ef[e,f] * W_edge[f, m*16+h] * h_s[e,h]
//            + sum_h    b_edge[m*16+h]              * h_s[e,h]
// Formulated as [16e x 272] @ [272 x 16m] GEMM per tile via 68 WMMA k-slices.
// A built on the fly: g[e,(f,h)] = ef[e,f]*h_s[e,h]; rows 256..271 use ef=1
// (bias). B (re-laid-out, bank-conflict-free pairs) staged once in LDS.
// Output tile is in C/D layout -> direct full-wave f32 atomic scatter.
// ---------------------------------------------------------------------------
#define EDGE_BLOCK 256
#define EDGE_WAVES (EDGE_BLOCK / 32)
#define KTOT 272                     // 256 bilinear rows + 16 bias rows
#define KSLICES (KTOT / 4)           // 68

__global__ __launch_bounds__(EDGE_BLOCK) void edge_msg_kernel(
    const float* __restrict__ ef, const float* __restrict__ We,
    const float* __restrict__ be, const float* __restrict__ hidden,
    const int* __restrict__ senders, const int* __restrict__ recv,
    float* __restrict__ messages) {
  // sW2[k2*32 + m*2 + parity] = W2[2*k2+parity, m]
  //   W2[k,m] = (k<256) ? We[(k>>4)*256 + m*16 + (k&15)] : be[m*16 + (k-256)]
  __shared__ float sW2[KTOT * 16];
  for (int i = threadIdx.x; i < KTOT * 16; i += EDGE_BLOCK) {
    const int k2 = i >> 5;
    const int rem = i & 31;
    const int m = rem >> 1;
    const int k = 2 * k2 + (rem & 1);
    sW2[i] = (k < 256) ? We[(k >> 4) * (MDIM * HDIM) + m * HDIM + (k & 15)]
                       : be[m * HDIM + (k & 15)];
  }
  __syncthreads();

  const unsigned lane = threadIdx.x & 31u;
  const unsigned col  = lane & 15u;   // m (B col) == edge-in-tile (A row)
  const unsigned half = lane >> 4;
  const int wave = threadIdx.x >> 5;
  const int gw = blockIdx.x * EDGE_WAVES + wave;
  const int nw = gridDim.x * EDGE_WAVES;
  const int ntiles = N_EDGES / 16;    // 25000 exact

  for (int tile = gw; tile < ntiles; tile += nw) {
    const int e0 = tile * 16;
    const int er = e0 + (int)col;

    // prefetch next tile's edge-feature row (uniform branch: EXEC untouched)
    if (tile + nw < ntiles) {
      __builtin_prefetch(ef + (size_t)(er + nw * 16) * FE, 0, 3);
    }

    // edge feature row for this lane's A-row
    float efr[FE];
#pragma unroll
    for (int j = 0; j < FE / 4; ++j) {
      float4 v = *(const float4*)(ef + (size_t)er * FE + 4 * j);
      efr[4 * j + 0] = v.x; efr[4 * j + 1] = v.y;
      efr[4 * j + 2] = v.z; efr[4 * j + 3] = v.w;
    }
    // gathered sender hidden, pre-offset by this lane's K sub-offset (2*half)
    const int s = senders[er];
    const float* hb = hidden + (size_t)s * HDIM + 2 * half;
    float2 hq[4];
#pragma unroll
    for (int q = 0; q < 4; ++q) hq[q] = *(const float2*)(hb + 4 * q);

    v8f c = {};
#pragma unroll
    for (int ks = 0; ks < KSLICES; ++ks) {
      const float2 hv = hq[ks & 3];               // h[4*(ks&3)+2*half +0/+1]
      const float fe = (ks < 64) ? efr[ks >> 2] : 1.0f;  // bias rows: ef=1
      v2f a, b;
      a.x = fe * hv.x;
      a.y = fe * hv.y;
      const float2 bw =
          *(const float2*)&sW2[(unsigned)(2 * ks + half) * 32u + col * 2u];
      b.x = bw.x;
      b.y = bw.y;
      c = wmma_f32_4(a, b, c);
    }

    // segment-sum scatter: C/D layout row = r + 8*half, col = m
#pragma unroll
    for (int r = 0; r < 8; ++r) {
      const int row = r + 8 * (int)half;
      const int rv = recv[e0 + row];
      atomicAdd(messages + (size_t)rv * MDIM + col, c[r]);
    }
  }
}

// ---------------------------------------------------------------------------
// Kernel 3: GRU over T=32 scalar inputs (seq = [hidden | messages]).
// One node per thread; weights staged in LDS (uniform broadcast reads).
// Per-thread sequence staged in LDS with stride-33 pad (33*tid mod 64 is a
// permutation over a wave's 32 lanes -> conflict-free ds_load per step),
// avoiding a dynamically-indexed private array (no scratch / select chains).
// Keras reset_after=True, gate order z|r|h. In-place hidden update.
// ---------------------------------------------------------------------------
#define SEQ_STRIDE 33

__global__ __launch_bounds__(256) void gru_kernel(
    float* __restrict__ hidden, const float* __restrict__ messages,
    const float* __restrict__ Wig, const float* __restrict__ Whg,
    const float* __restrict__ big, const float* __restrict__ bhg) {
  __shared__ float sWh[HDIM * 3 * HDIM];  // 768
  __shared__ float sWi[3 * HDIM], sbi[3 * HDIM], sbh[3 * HDIM];
  __shared__ float sseq[256 * SEQ_STRIDE];
  for (int i = threadIdx.x; i < HDIM * 3 * HDIM; i += 256) sWh[i] = Whg[i];
  if (threadIdx.x < 3 * HDIM) {
    sWi[threadIdx.x] = Wig[threadIdx.x];
    sbi[threadIdx.x] = big[threadIdx.x];
    sbh[threadIdx.x] = bhg[threadIdx.x];
  }
  __syncthreads();

  const int n = blockIdx.x * 256 + threadIdx.x;
  if (n >= N_NODES) return;

  float* myseq = &sseq[threadIdx.x * SEQ_STRIDE];
#pragma unroll
  for (int j = 0; j < HDIM / 4; ++j) {
    float4 v = *(const float4*)(hidden + (size_t)n * HDIM + 4 * j);
    myseq[4 * j + 0] = v.x; myseq[4 * j + 1] = v.y;
    myseq[4 * j + 2] = v.z; myseq[4 * j + 3] = v.w;
  }
#pragma unroll
  for (int j = 0; j < MDIM / 4; ++j) {
    float4 v = *(const float4*)(messages + (size_t)n * MDIM + 4 * j);
    myseq[HDIM + 4 * j + 0] = v.x; myseq[HDIM + 4 * j + 1] = v.y;
    myseq[HDIM + 4 * j + 2] = v.z; myseq[HDIM + 4 * j + 3] = v.w;
  }

  float h[HDIM];
#pragma unroll
  for (int j = 0; j < HDIM; ++j) h[j] = 0.0f;

#pragma unroll 1
  for (int t = 0; t < HDIM + MDIM; ++t) {
    const float x = myseq[t];
    float hn[HDIM];
#pragma unroll
    for (int j = 0; j < HDIM; ++j) {
      float hz = sbh[j], hr = sbh[HDIM + j], hh = sbh[2 * HDIM + j];
#pragma unroll
      for (int k = 0; k < HDIM; ++k) {
        const float hk = h[k];
        hz = fmaf(hk, sWh[k * 3 * HDIM + j], hz);
        hr = fmaf(hk, sWh[k * 3 * HDIM + HDIM + j], hr);
        hh = fmaf(hk, sWh[k * 3 * HDIM + 2 * HDIM + j], hh);
      }
      const float xz = fmaf(x, sWi[j], sbi[j]);
      const float xr = fmaf(x, sWi[HDIM + j], sbi[HDIM + j]);
      const float xh = fmaf(x, sWi[2 * HDIM + j], sbi[2 * HDIM + j]);
      const float z = 1.0f / (1.0f + __expf(-(xz + hz)));
      const float r = 1.0f / (1.0f + __expf(-(xr + hr)));
      const float pre = xh + r * hh;
      const float e2 = __expf(-2.0f * pre);
      const float hc = (1.0f - e2) / (1.0f + e2);
      hn[j] = z * h[j] + (1.0f - z) * hc;
    }
#pragma unroll
    for (int j = 0; j < HDIM; ++j) h[j] = hn[j];
  }

#pragma unroll
  for (int j = 0; j < HDIM / 4; ++j) {
    float4 v;
    v.x = h[4 * j + 0]; v.y = h[4 * j + 1];
    v.z = h[4 * j + 2]; v.w = h[4 * j + 3];
    *(float4*)(hidden + (size_t)n * HDIM + 4 * j) = v;
  }
}

// ---------------------------------------------------------------------------
// Kernel 4: readout  out[0] = sum_n ( [h|h0]@W_ri + b_ri ) * ( h@W_rj + b_rj )
// ---------------------------------------------------------------------------
__global__ __launch_bounds__(256) void readout_kernel(
    const float* __restrict__ hidden, const float* __restrict__ hidden0,
    const float* __restrict__ Wri, const float* __restrict__ bri,
    const float* __restrict__ Wrj, const float* __restrict__ brj,
    float* __restrict__ out) {
  __shared__ float red[256];
  const int n = blockIdx.x * 256 + threadIdx.x;
  float val = 0.0f;
  if (n < N_NODES) {
    float ia = bri[0], ja = brj[0];
#pragma unroll
    for (int k = 0; k < HDIM; ++k) {
      const float hv = hidden[(size_t)n * HDIM + k];
      ia = fmaf(hv, Wri[k], ia);
      ja = fmaf(hv, Wrj[k], ja);
      ia = fmaf(hidden0[(size_t)n * HDIM + k], Wri[HDIM + k], ia);
    }
    val = ia * ja;
  }
  red[threadIdx.x] = val;
  __syncthreads();
  for (int s = 128; s > 0; s >>= 1) {
    if (threadIdx.x < s) red[threadIdx.x] += red[threadIdx.x + s];
    __syncthreads();
  }
  if (threadIdx.x == 0) atomicAdd(out, red[0]);
}

// ---------------------------------------------------------------------------
extern "C" void kernel_launch(void* const* d_in, const int* in_sizes, int n_in,
                              void* d_out, int out_size, void* d_ws,
                              size_t ws_size, hipStream_t stream) {
  (void)in_sizes; (void)n_in; (void)out_size; (void)ws_size;
  const float* nf    = (const float*)d_in[0];
  const float* efeat = (const float*)d_in[1];
  const float* Wini  = (const float*)d_in[2];
  const float* bini  = (const float*)d_in[3];
  const float* We    = (const float*)d_in[4];
  const float* be    = (const float*)d_in[5];
  const float* Wig   = (const float*)d_in[6];
  const float* Whg   = (const float*)d_in[7];
  const float* big   = (const float*)d_in[8];
  const float* bhg   = (const float*)d_in[9];
  const float* Wri   = (const float*)d_in[10];
  const float* bri   = (const float*)d_in[11];
  const float* Wrj   = (const float*)d_in[12];
  const float* brj   = (const float*)d_in[13];
  const int*   recv  = (const int*)d_in[14];
  const int*   send  = (const int*)d_in[15];

  float* ws       = (float*)d_ws;
  float* hidden   = ws;                          // N*H
  float* hidden0  = ws + (size_t)N_NODES * HDIM; // N*H
  float* messages = ws + (size_t)2 * N_NODES * HDIM;
  float* outf     = (float*)d_out;

  zero_kernel<<<1, 1, 0, stream>>>(outf, 1);

  const int init_tiles = (N_NODES + 15) / 16;
  init_hidden_kernel<<<init_tiles, 32, 0, stream>>>(nf, Wini, bini, hidden,
                                                    hidden0);

  const int msg_elems = N_NODES * MDIM;
  for (int it = 0; it < N_ITERS; ++it) {
    zero_kernel<<<(msg_elems + 255) / 256, 256, 0, stream>>>(messages,
                                                             msg_elems);
    edge_msg_kernel<<<320, EDGE_BLOCK, 0, stream>>>(efeat, We, be, hidden,
                                                    send, recv, messages);
    gru_kernel<<<(N_NODES + 255) / 256, 256, 0, stream>>>(hidden, messages,
                                                          Wig, Whg, big, bhg);
  }

  readout_kernel<<<(N_NODES + 255) / 256, 256, 0, stream>>>(
      hidden, hidden0, Wri, bri, Wrj, brj, outf);
}